// EstimatorQNN_65481071407187
// MI455X (gfx1250) — compile-verified
//
#include <hip/hip_runtime.h>
#include <hip/hip_bf16.h>
#include <math.h>

#define NQ  10
#define DIM 1024
#define NL  4

typedef __attribute__((ext_vector_type(2))) float v2f;
typedef __attribute__((ext_vector_type(8))) float v8f;

// ---------------------------------------------------------------------------
// Prologue: build per-layer 32x32 complex operators.
//   A  = kron(G_w, w=0..4)   (acts on index bits 9..5; wire w <-> bit 4-w of row)
//   Bt = B^T, B = kron(G_w, w=5..9)  (bits 4..0; wire w <-> bit 9-w of col)
// ws layout per layer l (4096 floats): [Ar | Ai | Btr | Bti], each 1024 f32.
// Total ws use: 4*4096*4 = 64 KB.
// ---------------------------------------------------------------------------
__global__ void build_layer_ops(const float* __restrict__ weights,
                                float* __restrict__ ws) {
    const int l   = blockIdx.x;   // layer
    const int tid = threadIdx.x;  // 256 threads
    __shared__ float Gr[NQ][2][2];
    __shared__ float Gi[NQ][2][2];
    if (tid < NQ) {
        const float phi   = weights[(l * NQ + tid) * 3 + 0];
        const float theta = weights[(l * NQ + tid) * 3 + 1];
        const float omega = weights[(l * NQ + tid) * 3 + 2];
        const float c  = cosf(0.5f * theta), s = sinf(0.5f * theta);
        const float ap = 0.5f * (phi + omega), am = 0.5f * (phi - omega);
        const float cp = cosf(ap), sp = sinf(ap);
        const float cm = cosf(am), sm = sinf(am);
        // Rot = RZ(omega) RY(theta) RZ(phi)
        Gr[tid][0][0] =  cp * c;  Gi[tid][0][0] = -sp * c;  // e^{-i ap} c
        Gr[tid][0][1] = -cm * s;  Gi[tid][0][1] = -sm * s;  // -e^{+i am} s
        Gr[tid][1][0] =  cm * s;  Gi[tid][1][0] = -sm * s;  // e^{-i am} s
        Gr[tid][1][1] =  cp * c;  Gi[tid][1][1] =  sp * c;  // e^{+i ap} c
    }
    __syncthreads();

    float* Ar  = ws + l * 4096;
    float* Ai  = Ar + 1024;
    float* Btr = Ar + 2048;
    float* Bti = Ar + 3072;

    for (int e = 0; e < 4; ++e) {
        const int idx = tid * 4 + e;
        {   // A[r][rp]
            const int r = idx >> 5, rp = idx & 31;
            float re = 1.f, im = 0.f;
            #pragma unroll
            for (int w = 0; w < 5; ++w) {
                const int bi = (r  >> (4 - w)) & 1;
                const int bj = (rp >> (4 - w)) & 1;
                const float gr = Gr[w][bi][bj], gi = Gi[w][bi][bj];
                const float nre = re * gr - im * gi;
                const float nim = re * gi + im * gr;
                re = nre; im = nim;
            }
            Ar[idx] = re; Ai[idx] = im;
        }
        {   // Bt[cp][c] = B[c][cp]
            const int cp = idx >> 5, c = idx & 31;
            float re = 1.f, im = 0.f;
            #pragma unroll
            for (int w = 5; w < 10; ++w) {
                const int bi = (c  >> (9 - w)) & 1;
                const int bj = (cp >> (9 - w)) & 1;
                const float gr = Gr[w][bi][bj], gi = Gi[w][bi][bj];
                const float nre = re * gr - im * gi;
                const float nim = re * gi + im * gr;
                re = nre; im = nim;
            }
            Btr[idx] = re; Bti[idx] = im;
        }
    }
}

// ---- WMMA f32 16x16x4 fragment loads (ISA 7.12.2 layouts) ------------------
// A (16x4): lane L holds row m=L&15; lanes 0-15 -> K={kb,kb+1}, 16-31 -> K={kb+2,kb+3}
__device__ __forceinline__ v2f loadA(const float* M, int rowBase, int kb, int lane) {
    const int m = rowBase + (lane & 15);
    const int k = kb + ((lane >> 4) << 1);
    v2f a; a.x = M[m * 32 + k]; a.y = M[m * 32 + k + 1];
    return a;
}
// B (4x16): lane L holds col n=L&15; lanes 0-15 -> K={kb,kb+1}, 16-31 -> K={kb+2,kb+3}
__device__ __forceinline__ v2f loadB(const float* M, int kb, int colBase, int lane) {
    const int n = colBase + (lane & 15);
    const int k = kb + ((lane >> 4) << 1);
    v2f b; b.x = M[k * 32 + n]; b.y = M[(k + 1) * 32 + n];
    return b;
}

#define CWMMA(accR, accI)                                                              \
    do {                                                                               \
        v2f an; an.x = -ai.x; an.y = -ai.y;                                            \
        accR = __builtin_amdgcn_wmma_f32_16x16x4_f32(false, ar, false, br, (short)0,   \
                                                     accR, false, false);              \
        accR = __builtin_amdgcn_wmma_f32_16x16x4_f32(false, an, false, bi, (short)0,   \
                                                     accR, false, false);              \
        accI = __builtin_amdgcn_wmma_f32_16x16x4_f32(false, ar, false, bi, (short)0,   \
                                                     accI, false, false);              \
        accI = __builtin_amdgcn_wmma_f32_16x16x4_f32(false, ai, false, br, (short)0,   \
                                                     accI, false, false);              \
    } while (0)

// ---------------------------------------------------------------------------
// Main: one 128-thread block (4 wave32) per sample. State = 32x32 complex f32
// in LDS (row = bits 9..5, col = bits 4..0). Per layer: S <- A*S*B^T via WMMA,
// then the CNOT-ring permutation as an LDS gather using the GF(2) inverse map.
// ---------------------------------------------------------------------------
__global__ void __launch_bounds__(128)
qsim_kernel(const float* __restrict__ inputs,
            const float* __restrict__ ws,
            float* __restrict__ out) {
    __shared__ float Sr[DIM], Si[DIM], Pr[DIM], Pi[DIM];
    const int b    = blockIdx.x;
    const int tid  = threadIdx.x;
    const int lane = tid & 31;
    const int wv   = tid >> 5;            // wave id 0..3
    const int ti   = (wv >> 1) * 16;      // output tile row base
    const int tj   = (wv & 1) * 16;       // output tile col base

    // ---- RX(pi*x) encoding: closed-form product state ----------------------
    float cs[NQ], sn[NQ];
    #pragma unroll
    for (int w = 0; w < NQ; ++w) {
        const float half_ang = 0.5f * 3.14159265358979323846f * inputs[b * NQ + w];
        cs[w] = cosf(half_ang); sn[w] = sinf(half_ang);
    }
    #pragma unroll
    for (int j = 0; j < 8; ++j) {
        const int x = tid * 8 + j;
        float mag = 1.f;
        #pragma unroll
        for (int w = 0; w < NQ; ++w)
            mag *= ((x >> (9 - w)) & 1) ? sn[w] : cs[w];
        const int k = __popc(x) & 3;      // amp = mag * (-i)^k
        float re = 0.f, im = 0.f;
        if      (k == 0) re =  mag;
        else if (k == 1) im = -mag;
        else if (k == 2) re = -mag;
        else             im =  mag;
        Sr[x] = re; Si[x] = im;
    }
    __syncthreads();

    for (int l = 0; l < NL; ++l) {
        const float* Ar  = ws + l * 4096;
        const float* Ai_ = Ar + 1024;
        const float* Btr = Ar + 2048;
        const float* Bti = Ar + 3072;

        // stage 1: P = A * S   (complex 32x32 matmul, per-wave 16x16 tile)
        {
            v8f accR = {}; v8f accI = {};
            #pragma unroll
            for (int k4 = 0; k4 < 8; ++k4) {
                const int kb = k4 * 4;
                const v2f ar = loadA(Ar,  ti, kb, lane);
                const v2f ai = loadA(Ai_, ti, kb, lane);
                const v2f br = loadB(Sr, kb, tj, lane);
                const v2f bi = loadB(Si, kb, tj, lane);
                CWMMA(accR, accI);
            }
            #pragma unroll
            for (int i = 0; i < 8; ++i) {   // C/D: VGPR i -> M = i + 8*(lane>=16)
                const int row = ti + i + ((lane >> 4) << 3);
                const int col = tj + (lane & 15);
                Pr[row * 32 + col] = accR[i];
                Pi[row * 32 + col] = accI[i];
            }
        }
        __syncthreads();

        // stage 2: S = P * B^T
        {
            v8f accR = {}; v8f accI = {};
            #pragma unroll
            for (int k4 = 0; k4 < 8; ++k4) {
                const int kb = k4 * 4;
                const v2f ar = loadA(Pr, ti, kb, lane);
                const v2f ai = loadA(Pi, ti, kb, lane);
                const v2f br = loadB(Btr, kb, tj, lane);
                const v2f bi = loadB(Bti, kb, tj, lane);
                CWMMA(accR, accI);
            }
            #pragma unroll
            for (int i = 0; i < 8; ++i) {
                const int row = ti + i + ((lane >> 4) << 3);
                const int col = tj + (lane & 15);
                Sr[row * 32 + col] = accR[i];
                Si[row * 32 + col] = accI[i];
            }
        }
        __syncthreads();

        // CNOT ring, range r = l+1: new[y] = old[f0(f1(...f9(y)))]
        {
            const int rr = l + 1;
            float tr[8], tm[8];
            #pragma unroll
            for (int j = 0; j < 8; ++j) {
                const int y = tid * 8 + j;
                int s = y;
                #pragma unroll
                for (int w = 9; w >= 0; --w) {
                    const int pc = 9 - w;
                    const int pt = 9 - ((w + rr) % 10);
                    s ^= ((s >> pc) & 1) << pt;
                }
                tr[j] = Sr[s]; tm[j] = Si[s];
            }
            __syncthreads();
            #pragma unroll
            for (int j = 0; j < 8; ++j) {
                const int y = tid * 8 + j;
                Sr[y] = tr[j]; Si[y] = tm[j];
            }
        }
        __syncthreads();
    }

    // ---- <Z> on last wire (LSB of index) -----------------------------------
    float local = 0.f;
    #pragma unroll
    for (int j = 0; j < 8; ++j) {
        const int y = tid * 8 + j;
        const float re = Sr[y], im = Si[y];
        local += ((y & 1) ? -1.f : 1.f) * (re * re + im * im);
    }
    Pr[tid] = local;
    __syncthreads();
    for (int stride = 64; stride > 0; stride >>= 1) {
        if (tid < stride) Pr[tid] += Pr[tid + stride];
        __syncthreads();
    }
    if (tid == 0) out[b] = Pr[0];
}

extern "C" void kernel_launch(void* const* d_in, const int* in_sizes, int n_in,
                              void* d_out, int out_size, void* d_ws, size_t ws_size,
                              hipStream_t stream) {
    const float* inputs  = (const float*)d_in[0];   // (B, 10) f32
    const float* weights = (const float*)d_in[1];   // (4, 10, 3) f32
    float* out = (float*)d_out;                     // (B, 1) f32
    float* ws  = (float*)d_ws;                      // needs 64 KB
    const int B = in_sizes[0] / NQ;

    build_layer_ops<<<NL, 256, 0, stream>>>(weights, ws);
    qsim_kernel<<<B, 128, 0, stream>>>(inputs, ws, out);
}